// ForwardLSTM_89129161327048
// MI455X (gfx1250) — compile-verified
//
#include <hip/hip_runtime.h>
#include <math.h>

// Single-token 2-layer LSTM step + vocab decoder for MI455X (gfx1250, wave32).
// Memory-bound GEMV (~680MB fp32 weights @ 23.3TB/s ~= 29us floor); dot products
// run through v_wmma_f32_16x16x4_f32 (16-row tiles per wave, x broadcast into all
// B columns so every D column equals A@x). Weights and x are loaded as b128; the
// K-slot permutation inside each WMMA is matched between A and B (valid: WMMA's K
// dimension is an unordered dot-product reduction).
//
// Concurrency: HBM latency*BW needs ~12MB in flight. The gates GEMV only has 512
// row-tiles, so it is K-split x4 (2048 waves/layer) with deterministic partials
// in scratch, reduced in the activation kernel. Decoder has 3125 tiles already.

typedef __attribute__((ext_vector_type(2))) float v2f;
typedef __attribute__((ext_vector_type(4))) float v4f;
typedef __attribute__((ext_vector_type(8))) float v8f;

constexpr int V  = 50000;
constexpr int H  = 2048;
constexpr int L  = 2;
constexpr int G4 = 4 * H;            // gate rows per layer (8192)
constexpr int WAVES_PER_BLOCK = 8;   // 256 threads
constexpr int KSPLIT = 4;            // K-chunks for gates GEMV
constexpr int KCH = H / KSPLIT;      // 512

__device__ __forceinline__ v8f wmma_k4(v2f a, v2f b, v8f c) {
  // D = A(16x4 f32) x B(4x16 f32) + C(16x16 f32)
  return __builtin_amdgcn_wmma_f32_16x16x4_f32(false, a, false, b,
                                               (short)0, c, false, false);
}

// wrow = &W[row][koff], xoff = &x[koff] with koff = (lane>=16 ? 4 : 0).
// Accumulates dot(W[row], x) over K into two 16x16 accumulators.
// Per 8-K step: one b128 NT weight load + one b128 x load + two WMMAs.
__device__ __forceinline__ void gemv_tile_accum(const float* __restrict__ wrow,
                                                const float* __restrict__ xoff,
                                                int K, v8f& c0, v8f& c1) {
#pragma unroll 8
  for (int k = 0; k < K; k += 8) {
    v4f w = __builtin_nontemporal_load((const v4f*)(wrow + k)); // streaming weights
    v4f x = *(const v4f*)(xoff + k);                            // cached x broadcast
    // WMMA #1: K slots = global {k, k+1} (lanes 0-15) / {k+4, k+5} (lanes 16-31)
    c0 = wmma_k4(w.lo, x.lo, c0);
    // WMMA #2: K slots = global {k+2, k+3} / {k+6, k+7}
    c1 = wmma_k4(w.hi, x.hi, c1);
  }
}

// part[kc][0:4H] partial of (W_ih @ x + W_hh @ h_prev) over K chunk kc.
// One 16-row tile x one K chunk per wave; 2048 waves total.
__global__ void lstm_gates_wmma(const float* __restrict__ Wi,
                                const float* __restrict__ Wh,
                                const float* __restrict__ xbase,
                                const int*   __restrict__ token, // nullptr -> xbase is x
                                const float* __restrict__ hprev,
                                float* __restrict__ part) {      // [KSPLIT][G4]
  const int wave = (int)threadIdx.x >> 5;
  const int wid  = (int)blockIdx.x * WAVES_PER_BLOCK + wave;
  if (wid >= (G4 / 16) * KSPLIT) return;     // whole-wave guard: EXEC stays all-1s
  const int tile = wid >> 2;                 // row tile
  const int kc   = wid & (KSPLIT - 1);       // K chunk
  const int lane = (int)threadIdx.x & 31;
  const int base = tile * 16;
  const int row  = base + (lane & 15);
  const int koff = (lane >> 4) << 2;         // lanes 0-15 -> k+0..3, 16-31 -> k+4..7
  const int k0   = kc * KCH;

  const float* x = xbase;
  if (token) x += (size_t)token[0] * H;      // layer 0: embedding row gather

  v8f c0 = {}, c1 = {};
  gemv_tile_accum(Wi + (size_t)row * H + k0 + koff, x     + k0 + koff, KCH, c0, c1);
  gemv_tile_accum(Wh + (size_t)row * H + k0 + koff, hprev + k0 + koff, KCH, c0, c1);
  v8f c = c0 + c1;

  if ((lane & 15) == 0) {                    // lane 0 -> rows base..+7, lane 16 -> +8..+15
    const int rb = base + ((lane >> 4) << 3);
    float* p = part + (size_t)kc * G4;
#pragma unroll
    for (int v = 0; v < 8; ++v)
      p[rb + v] = c[v];                      // each slot written exactly once
  }
}

// Reduce K-split partials + biases, then i,f,g,o activations + state update.
__global__ void lstm_activate(const float* __restrict__ part,  // [KSPLIT][G4]
                              const float* __restrict__ bi,
                              const float* __restrict__ bh,
                              const float* __restrict__ cprev,
                              float* __restrict__ hout,
                              float* __restrict__ cout) {
  const int j = (int)blockIdx.x * (int)blockDim.x + (int)threadIdx.x;
  if (j >= H) return;
  float gate[4];
#pragma unroll
  for (int q = 0; q < 4; ++q) {
    const int r = j + q * H;
    float s = bi[r] + bh[r];
#pragma unroll
    for (int kc = 0; kc < KSPLIT; ++kc) s += part[(size_t)kc * G4 + r];
    gate[q] = s;
  }
  const float i = 1.0f / (1.0f + expf(-gate[0]));
  const float f = 1.0f / (1.0f + expf(-gate[1]));
  const float g = tanhf(gate[2]);
  const float o = 1.0f / (1.0f + expf(-gate[3]));
  const float c = f * cprev[j] + i * g;
  const float h = o * tanhf(c);
  cout[j] = c;
  hout[j] = h;
}

// out[0:V] = w_dec @ x + b_dec   (3125 waves, full K per wave)
__global__ void decoder_wmma(const float* __restrict__ W,
                             const float* __restrict__ bias,
                             const float* __restrict__ x,
                             float* __restrict__ out) {
  const int wave = (int)threadIdx.x >> 5;
  const int tile = (int)blockIdx.x * WAVES_PER_BLOCK + wave;
  if (tile >= V / 16) return;                // V = 3125 * 16
  const int lane = (int)threadIdx.x & 31;
  const int base = tile * 16;
  const int row  = base + (lane & 15);
  const int koff = (lane >> 4) << 2;

  v8f c0 = {}, c1 = {};
  gemv_tile_accum(W + (size_t)row * H + koff, x + koff, H, c0, c1);
  v8f c = c0 + c1;

  if ((lane & 15) == 0) {
    const int rb = base + ((lane >> 4) << 3);
#pragma unroll
    for (int v = 0; v < 8; ++v)
      out[rb + v] = c[v] + bias[rb + v];
  }
}

extern "C" void kernel_launch(void* const* d_in, const int* in_sizes, int n_in,
                              void* d_out, int out_size, void* d_ws, size_t ws_size,
                              hipStream_t stream) {
  const int*   token   = (const int*)  d_in[0];
  const float* prev_hx = (const float*)d_in[1];  // [L,H]
  const float* prev_cx = (const float*)d_in[2];  // [L,H]
  const float* encoder = (const float*)d_in[3];  // [V,H]
  const float* W_ih    = (const float*)d_in[4];  // [L,4H,H]
  const float* W_hh    = (const float*)d_in[5];  // [L,4H,H]
  const float* b_ih    = (const float*)d_in[6];  // [L,4H]
  const float* b_hh    = (const float*)d_in[7];  // [L,4H]
  const float* w_dec   = (const float*)d_in[8];  // [V,H]
  const float* b_dec   = (const float*)d_in[9];  // [V]

  float* out    = (float*)d_out;          // [V] logits
  float* hx_out = out + V;                // [L,H]
  float* cx_out = hx_out + (size_t)L * H; // [L,H]
  float* part   = (float*)d_ws;           // [KSPLIT][G4] partials (128KB), reused

  const dim3 blk(32 * WAVES_PER_BLOCK);
  const int gates_blocks = ((G4 / 16) * KSPLIT + WAVES_PER_BLOCK - 1) / WAVES_PER_BLOCK; // 256
  const int act_blocks   = (H + 255) / 256;                                              // 8
  const int dec_blocks   = (V / 16 + WAVES_PER_BLOCK - 1) / WAVES_PER_BLOCK;             // 391

  // Layer 0: x = encoder[token]
  lstm_gates_wmma<<<gates_blocks, blk, 0, stream>>>(
      W_ih, W_hh, encoder, token, prev_hx, part);
  lstm_activate<<<act_blocks, blk, 0, stream>>>(part, b_ih, b_hh, prev_cx,
                                                hx_out, cx_out);

  // Layer 1: x = h0 (just computed)
  lstm_gates_wmma<<<gates_blocks, blk, 0, stream>>>(
      W_ih + (size_t)G4 * H, W_hh + (size_t)G4 * H, hx_out, nullptr,
      prev_hx + H, part);
  lstm_activate<<<act_blocks, blk, 0, stream>>>(part, b_ih + G4, b_hh + G4,
                                                prev_cx + H, hx_out + H, cx_out + H);

  // Decoder: out = w_dec @ h1 + b_dec
  decoder_wmma<<<dec_blocks, blk, 0, stream>>>(w_dec, b_dec, hx_out + H, out);
}